// PotentialModel_42004780155431
// MI455X (gfx1250) — compile-verified
//
#include <hip/hip_runtime.h>
#include <math.h>

#define NA    4096
#define NTB   4000    // bonds
#define NTANG 8000    // angles
#define NTAC  4000    // charmm angles
#define NTD   12000   // dihedrals
#define NTDC  6000    // charmm dihedrals
#define NTI   2000    // impropers
#define TOTB  (NTB+NTANG+NTAC+NTD+NTDC+NTI)   // 36000
#define EPSF      1e-12f
#define COULOMBF  332.33f

typedef __attribute__((ext_vector_type(2))) float v2f;
typedef __attribute__((ext_vector_type(8))) float v8f;

struct F3 { float x, y, z; };
__device__ __forceinline__ F3 fsub(F3 a, F3 b){ return {a.x-b.x, a.y-b.y, a.z-b.z}; }
__device__ __forceinline__ float fdot(F3 a, F3 b){ return a.x*b.x + a.y*b.y + a.z*b.z; }
__device__ __forceinline__ F3 fcross(F3 a, F3 b){
  return {a.y*b.z - a.z*b.y, a.z*b.x - a.x*b.z, a.x*b.y - a.y*b.x};
}
__device__ __forceinline__ F3 fscale(F3 a, float s){ return {a.x*s, a.y*s, a.z*s}; }
__device__ __forceinline__ F3 loadp(const float* __restrict__ p, int i){
  return {p[3*i], p[3*i+1], p[3*i+2]};
}

__device__ __forceinline__ float angle3(F3 p0, F3 p1, F3 p2){
  F3 u = fsub(p0, p1), v = fsub(p2, p1);
  float c = fdot(u, v) * rsqrtf(fdot(u, u) * fdot(v, v) + EPSF);
  c = fminf(1.0f - 1e-7f, fmaxf(-1.0f + 1e-7f, c));
  return acosf(c);
}

__device__ __forceinline__ float dihedral4(F3 p0, F3 p1, F3 p2, F3 p3){
  F3 b1 = fsub(p1, p0), b2 = fsub(p2, p1), b3 = fsub(p3, p2);
  F3 n1 = fcross(b1, b2), n2 = fcross(b2, b3);
  F3 b2n = fscale(b2, rsqrtf(fdot(b2, b2) + EPSF));
  return atan2f(fdot(fcross(n1, b2n), n2), fdot(n1, n2));
}

// LJ + Coulomb for one pair, given squared distance and combined parameters.
__device__ __forceinline__ float pair_energy(float r2, float epsij, float sgij,
                                             float qq){
  float rinv  = rsqrtf(r2 + EPSF);
  float rinv2 = rinv * rinv;
  float sr2   = sgij * sgij * rinv2;
  float sr6   = sr2 * sr2 * sr2;
  return 4.0f * epsij * (sr6 * sr6 - sr6) + COULOMBF * qq * rinv;
}

__global__ void zero_acc_kernel(double* acc){ acc[0] = 0.0; }
__global__ void finalize_kernel(const double* __restrict__ acc, float* __restrict__ out){
  out[0] = (float)acc[0];
}

// ---------------- Bonded terms: one thread per bonded item ----------------
__global__ __launch_bounds__(256)
void bonded_kernel(const float* __restrict__ pos,
                   const float* __restrict__ bondC,  const float* __restrict__ angC,
                   const float* __restrict__ angcC,  const float* __restrict__ dihC,
                   const float* __restrict__ dihcC,  const float* __restrict__ impC,
                   const int* __restrict__ bondI,    const int* __restrict__ angI,
                   const int* __restrict__ angcI,    const int* __restrict__ dihI,
                   const int* __restrict__ dihcI,    const int* __restrict__ impI,
                   double* __restrict__ acc)
{
  int gid = blockIdx.x * 256 + threadIdx.x;
  double e = 0.0;
  if (gid < NTB) {
    const int* r = bondI + gid * 3;
    F3 d = fsub(loadp(pos, r[0]), loadp(pos, r[1]));
    int t = r[2];
    float rb = sqrtf(fdot(d, d) + EPSF);
    float dr = rb - bondC[2*t+1];
    e = (double)(bondC[2*t] * dr * dr);
  } else if (gid < NTB + NTANG) {
    const int* r = angI + (gid - NTB) * 4;
    float th = angle3(loadp(pos, r[0]), loadp(pos, r[1]), loadp(pos, r[2]));
    int t = r[3];
    float d = th - angC[2*t+1];
    e = (double)(angC[2*t] * d * d);
  } else if (gid < NTB + NTANG + NTAC) {
    const int* r = angcI + (gid - NTB - NTANG) * 4;
    F3 p0 = loadp(pos, r[0]), p1 = loadp(pos, r[1]), p2 = loadp(pos, r[2]);
    int t = r[3];
    float th = angle3(p0, p1, p2);
    F3 d02 = fsub(p0, p2);
    float r13 = sqrtf(fdot(d02, d02) + EPSF);
    float dth = th  - angcC[4*t+1];
    float dub = r13 - angcC[4*t+3];
    e = (double)(angcC[4*t] * dth * dth + angcC[4*t+2] * dub * dub);
  } else if (gid < NTB + NTANG + NTAC + NTD) {
    const int* r = dihI + (gid - NTB - NTANG - NTAC) * 5;
    float ph = dihedral4(loadp(pos, r[0]), loadp(pos, r[1]),
                         loadp(pos, r[2]), loadp(pos, r[3]));
    float c = cosf(ph);
    const float* A = dihC + 5 * r[4];
    e = (double)(A[0] + c*(A[1] + c*(A[2] + c*(A[3] + c*A[4]))));
  } else if (gid < NTB + NTANG + NTAC + NTD + NTDC) {
    const int* r = dihcI + (gid - NTB - NTANG - NTAC - NTD) * 5;
    float ph = dihedral4(loadp(pos, r[0]), loadp(pos, r[1]),
                         loadp(pos, r[2]), loadp(pos, r[3]));
    const float* C = dihcC + 4 * r[4];
    e = (double)(C[0] * (1.0f + cosf(C[1] * ph - C[2])));
  } else if (gid < TOTB) {
    const int* r = impI + (gid - NTB - NTANG - NTAC - NTD - NTDC) * 5;
    float ch = dihedral4(loadp(pos, r[0]), loadp(pos, r[1]),
                         loadp(pos, r[2]), loadp(pos, r[3]));
    const float* C = impC + 2 * r[4];
    float d = ch - C[1];
    e = (double)(C[0] * d * d);
  }
  #pragma unroll
  for (int off = 16; off; off >>= 1) e += __shfl_down(e, off, 32);
  if ((threadIdx.x & 31) == 0) atomicAdd(acc, e);
}

// ---------------- Non-bonded term: WMMA Gram-matrix distance kernel ----------------
// One wave32 per 16x16 atom tile-pair. r2[m][n] = |pi|^2 - 2 pi.pj + |pj|^2 where
// D = A x B with A(16x4) rows = [-2x,-2y,-2z,1] and B(4x16) cols = [x;y;z;|p|^2],
// computed by one V_WMMA_F32_16X16X4_F32; |pi|^2 added per element from LDS.
// After the (single, full-EXEC) WMMA, off-diagonal tile pairs (I<J, 99.2% of
// waves) take a fully unconditional straight-line loop; only the 256 diagonal
// tiles (I==J) use per-element selects for the i<j ordering.
struct TileAttr { float x[16], y[16], z[16], r2[16], q[16], se[16], sg[16]; };

__global__ __launch_bounds__(256)
void nb_kernel(const float* __restrict__ pos, const float* __restrict__ charge,
               const float* __restrict__ eps, const float* __restrict__ sig,
               const float* __restrict__ mask, double* __restrict__ acc)
{
  __shared__ TileAttr SI[8], SJ[8];
  const int w    = threadIdx.x >> 5;
  const int lane = threadIdx.x & 31;
  const int wid  = blockIdx.x * 8 + w;     // [0, 65536)
  const int I    = wid >> 8;               // tile row  [0, 256)
  const int J    = wid & 255;              // tile col  [0, 256)
  const bool active = (J >= I);
  const bool hi  = lane >= 16;
  const int  t   = lane & 15;

  if (active) {
    // Stage: lanes 0-15 load tile-I atom t, lanes 16-31 load tile-J atom t.
    int ga = (hi ? J : I) * 16 + t;
    TileAttr* S = hi ? &SJ[w] : &SI[w];
    float x = pos[3*ga]   - 40.0f;         // center box to reduce |p|^2 cancellation
    float y = pos[3*ga+1] - 40.0f;
    float z = pos[3*ga+2] - 40.0f;
    S->x[t]  = x;  S->y[t] = y;  S->z[t] = z;
    S->r2[t] = x*x + y*y + z*z;
    S->q[t]  = charge[ga];
    S->se[t] = sqrtf(eps[ga]);             // sqrt(ei*ej) = sqrt(ei)*sqrt(ej)
    S->sg[t] = sig[ga];
  }
  __syncthreads();

  double esum = 0.0;
  if (active) {
    // A-matrix 16x4 f32 layout: lane<16 holds (K0,K1) for M=lane; lane>=16 holds (K2,K3).
    // B-matrix 4x16 f32 layout: lane<16 holds (K0,K1) for N=lane; lane>=16 holds (K2,K3).
    v2f a, b;
    a[0] = hi ? (-2.0f * SI[w].z[t]) : (-2.0f * SI[w].x[t]);
    a[1] = hi ? 1.0f                 : (-2.0f * SI[w].y[t]);
    b[0] = hi ? SJ[w].z[t]           : SJ[w].x[t];
    b[1] = hi ? SJ[w].r2[t]          : SJ[w].y[t];
    v8f c = {};
    c = __builtin_amdgcn_wmma_f32_16x16x4_f32(
        /*neg_a=*/false, a, /*neg_b=*/false, b,
        /*c_mod=*/(short)0, c, /*reuse_a=*/false, /*reuse_b=*/false);

    // C/D layout: lanes 0-15 -> N=lane, VGPR v -> M=v; lanes 16-31 -> M=v+8.
    const int n     = t;
    const int mbase = hi ? 8 : 0;
    const int gj    = J * 16 + n;

    // Consume the whole accumulator immediately so the WMMA result is fully
    // dead before any branching below.
    float r2a[8];
    #pragma unroll
    for (int v = 0; v < 8; ++v) r2a[v] = c[v] + SI[w].r2[mbase + v];

    const float qj  = SJ[w].q[n];
    const float sej = SJ[w].se[n];
    const float sgj = SJ[w].sg[n];
    const size_t mrow = (size_t)gj;        // column index within row gi

    if (I < J) {
      // Hot path: every (gi,gj) satisfies gi<gj. No predicates anywhere, so
      // mask loads and math stay unconditional straight-line code.
      #pragma unroll
      for (int v = 0; v < 8; ++v) {
        int m    = mbase + v;
        int gi   = I * 16 + m;
        float mk = mask[((size_t)gi << 12) + mrow];
        float ep = pair_energy(r2a[v], SI[w].se[m] * sej,
                               0.5f * (SI[w].sg[m] + sgj),
                               SI[w].q[m] * qj);
        esum += (double)(mk * ep);
      }
    } else {
      // Diagonal tile (I==J): enforce i<j with VALU selects; clamp r2 for
      // invalid elements so (sig/r)^12 cannot produce inf*0=NaN.
      #pragma unroll
      for (int v = 0; v < 8; ++v) {
        int m      = mbase + v;
        int gi     = I * 16 + m;
        bool valid = gi < gj;
        float mk   = mask[((size_t)gi << 12) + mrow];
        float r2   = valid ? r2a[v] : 1.0f;
        float ep   = pair_energy(r2, SI[w].se[m] * sej,
                                 0.5f * (SI[w].sg[m] + sgj),
                                 SI[w].q[m] * qj);
        float wgt  = valid ? mk : 0.0f;
        esum += (double)(wgt * ep);
      }
    }
  }
  #pragma unroll
  for (int off = 16; off; off >>= 1) esum += __shfl_down(esum, off, 32);
  if (lane == 0 && active) atomicAdd(acc, esum);
}

extern "C" void kernel_launch(void* const* d_in, const int* in_sizes, int n_in,
                              void* d_out, int out_size, void* d_ws, size_t ws_size,
                              hipStream_t stream) {
  const float* pos   = (const float*)d_in[0];
  const float* q     = (const float*)d_in[1];
  const float* eps   = (const float*)d_in[2];
  const float* sig   = (const float*)d_in[3];
  const float* mask  = (const float*)d_in[4];
  const float* bondC = (const float*)d_in[5];
  const float* angC  = (const float*)d_in[6];
  const float* angcC = (const float*)d_in[7];
  const float* dihC  = (const float*)d_in[8];
  const float* dihcC = (const float*)d_in[9];
  const float* impC  = (const float*)d_in[10];
  const int* bondI   = (const int*)d_in[11];
  const int* angI    = (const int*)d_in[12];
  const int* angcI   = (const int*)d_in[13];
  const int* dihI    = (const int*)d_in[14];
  const int* dihcI   = (const int*)d_in[15];
  const int* impI    = (const int*)d_in[16];
  // d_in[17] (pair_idx) intentionally unused: pairs are regenerated as triu order.

  double* acc = (double*)d_ws;

  zero_acc_kernel<<<1, 1, 0, stream>>>(acc);
  bonded_kernel<<<(TOTB + 255) / 256, 256, 0, stream>>>(
      pos, bondC, angC, angcC, dihC, dihcC, impC,
      bondI, angI, angcI, dihI, dihcI, impI, acc);
  nb_kernel<<<(256 * 256) / 8, 256, 0, stream>>>(pos, q, eps, sig, mask, acc);
  finalize_kernel<<<1, 1, 0, stream>>>(acc, (float*)d_out);
}